// feat_prototype_distance_module_65901978190133
// MI455X (gfx1250) — compile-verified
//
#include <hip/hip_runtime.h>
#include <hip/hip_bf16.h>

typedef float v2f __attribute__((ext_vector_type(2)));
typedef float v8f __attribute__((ext_vector_type(8)));

#define C_DIM 2048
#define HW    4096
#define K_CLS 21

// ---------------------------------------------------------------------------
// Kernel 1: proto_norm[k] = sqrt(sum_c proto[k][c]^2), k = 0..20  -> d_ws
// One wave32 per class row; lane-strided partial sums + xor-shuffle reduce.
// ---------------------------------------------------------------------------
__global__ __launch_bounds__(672)
void proto_norm_kernel(const float* __restrict__ proto, float* __restrict__ pnorm) {
    const int warp = threadIdx.x >> 5;
    const int lane = threadIdx.x & 31;
    if (warp >= K_CLS) return;
    const float* row = proto + (size_t)warp * C_DIM;
    float s = 0.0f;
    #pragma unroll 4
    for (int c = lane; c < C_DIM; c += 32) {
        float v = row[c];
        s = fmaf(v, v, s);
    }
    #pragma unroll
    for (int off = 16; off > 0; off >>= 1)
        s += __shfl_xor(s, off, 32);
    if (lane == 0) pnorm[warp] = sqrtf(s);
}

// ---------------------------------------------------------------------------
// Kernel 2: fused GEMM (fp32 WMMA 16x16x4) + feat-norm + cosine + ReLU.
// One wave = one 16-pixel tile of one image, both 16-row M tiles (M padded
// to 32; rows >= 21 clamped to row 20 and never stored).
// ---------------------------------------------------------------------------
__global__ __launch_bounds__(256)
void cos_sim_wmma_kernel(const float* __restrict__ feat,
                         const float* __restrict__ proto,
                         const float* __restrict__ pnorm,
                         float* __restrict__ out) {
    const int lane = threadIdx.x & 31;
    const int wave = threadIdx.x >> 5;
    const int tile = blockIdx.x * 8 + wave;    // 0..2047  (256 tiles per image)
    const int n    = tile >> 8;                // image index
    const int p0   = (tile & 255) << 4;        // pixel base inside image
    const int col  = lane & 15;                // N index (pixel within tile)
    const int half = lane >> 4;                // 0 or 1: selects K pair

    // B source: lane loads feat[c + 2*half + {0,1}][p0 + col] each K-step.
    const float* bptr = feat + (size_t)n * C_DIM * HW
                             + (size_t)(2 * half) * HW + (p0 + col);

    // A source: row = M (lane%16 [+16 for tile1, clamped]), cols c + 2*half.
    int m1 = 16 + col; if (m1 > K_CLS - 1) m1 = K_CLS - 1;   // clamp pad rows
    const float* aptr0 = proto + (size_t)col * C_DIM + 2 * half;
    const float* aptr1 = proto + (size_t)m1  * C_DIM + 2 * half;

    v8f acc0 = {};
    v8f acc1 = {};
    float nrm = 0.0f;

    #pragma unroll 4
    for (int c = 0; c < C_DIM; c += 4) {
        // Streaming feat reads: touched exactly once -> non-temporal.
        float b0 = __builtin_nontemporal_load(bptr + (size_t)c * HW);
        float b1 = __builtin_nontemporal_load(bptr + (size_t)(c + 1) * HW);
        v2f b = {b0, b1};
        v2f a0 = *(const v2f*)(aptr0 + c);   // 8B aligned: c%4==0, half offset even
        v2f a1 = *(const v2f*)(aptr1 + c);

        nrm = fmaf(b0, b0, nrm);
        nrm = fmaf(b1, b1, nrm);

        // D = A(16x4) * B(4x16) + C   (fp32 WMMA, wave32)
        acc0 = __builtin_amdgcn_wmma_f32_16x16x4_f32(
                   false, a0, false, b, (short)0, acc0, false, false);
        acc1 = __builtin_amdgcn_wmma_f32_16x16x4_f32(
                   false, a1, false, b, (short)0, acc1, false, false);
    }

    // Per-pixel feat norm: lane-half pair covers complementary channel rows.
    nrm += __shfl_xor(nrm, 16, 32);
    const float fn = sqrtf(nrm);

    // Epilogue: C/D layout is VGPR j -> M = j + 8*(lane>=16), N = lane%16.
    float* outn = out + (size_t)n * K_CLS * HW + p0 + col;
    #pragma unroll
    for (int j = 0; j < 8; ++j) {
        const int m = j + (half << 3);                 // 0..15, always valid
        const float d0 = fmaxf(fn * pnorm[m], 1e-8f);
        outn[(size_t)m * HW] = fmaxf(acc0[j] / d0, 0.0f);

        const int mm = m + 16;                         // 16..31, valid if < 21
        if (mm < K_CLS) {
            const float d1 = fmaxf(fn * pnorm[mm], 1e-8f);
            outn[(size_t)mm * HW] = fmaxf(acc1[j] / d1, 0.0f);
        }
    }
}

// ---------------------------------------------------------------------------
extern "C" void kernel_launch(void* const* d_in, const int* in_sizes, int n_in,
                              void* d_out, int out_size, void* d_ws, size_t ws_size,
                              hipStream_t stream) {
    const float* feat  = (const float*)d_in[0];   // [8, 2048, 64, 64] fp32
    const float* proto = (const float*)d_in[1];   // [21, 2048] fp32
    // d_in[2] = class_numbers (fixed 21)
    float* pnorm = (float*)d_ws;                  // 21 floats
    float* out   = (float*)d_out;                 // [8, 21, 64, 64] fp32

    proto_norm_kernel<<<1, 672, 0, stream>>>(proto, pnorm);
    cos_sim_wmma_kernel<<<256, 256, 0, stream>>>(feat, proto, pnorm, out);
}